// HGATConv_17119739642017
// MI455X (gfx1250) — compile-verified
//
#include <hip/hip_runtime.h>

typedef __attribute__((ext_vector_type(2))) float v2f;
typedef __attribute__((ext_vector_type(8))) float v8f;

#define DIN  128
#define DOUT 128
#define HEDGES 20000   // num_hyperedges: fixed Python scalar in the reference

// ---------- float <-> order-preserving unsigned (for atomic float max) ----------
__device__ __forceinline__ unsigned ordf(float f) {
  unsigned u = __float_as_uint(f);
  return (u & 0x80000000u) ? ~u : (u | 0x80000000u);
}
__device__ __forceinline__ float unordf(unsigned u) {
  return __uint_as_float((u & 0x80000000u) ? (u & 0x7FFFFFFFu) : ~u);
}

// ---------- init: zero accumulation buffers + softmax stats ----------
__global__ void k_init(float* __restrict__ h_e, float* __restrict__ h_n,
                       unsigned* __restrict__ stats, long long he_count, long long hn_count) {
  long long tid = (long long)blockIdx.x * blockDim.x + threadIdx.x;
  long long stride = (long long)gridDim.x * blockDim.x;
  for (long long i = tid; i < hn_count; i += stride) h_n[i] = 0.0f;
  for (long long i = tid; i < he_count; i += stride) h_e[i] = 0.0f;
  if (tid == 0) { stats[0] = 0u; stats[1] = 0u; stats[2] = 0u; stats[3] = 0u; }
}

// ---------- GEMM: h = x @ W + b via V_WMMA_F32_16X16X4_F32 ----------
// One wave computes a 16-row x 128-col strip (8 accumulators), 32 k-steps of K=4.
__global__ void k_gemm_wmma(const float* __restrict__ x, const float* __restrict__ W,
                            const float* __restrict__ bias, float* __restrict__ h,
                            int nrows) {
  const int wave = blockIdx.x * (blockDim.x >> 5) + (threadIdx.x >> 5);
  const int lane = threadIdx.x & 31;
  const int half = lane >> 4;     // K-half selector for A/B fragments
  const int ln   = lane & 15;     // M (for A) / N (for B,C,D)
  const int row0 = wave * 16;
  if (row0 + 16 > nrows) return;

  v8f acc[8];
  #pragma unroll
  for (int c = 0; c < 8; ++c) {
    const float bv = bias[c * 16 + ln];     // C(m,n) = b[n] for all m
    #pragma unroll
    for (int i = 0; i < 8; ++i) acc[c][i] = bv;
  }

  for (int kk = 0; kk < DIN; kk += 4) {
    const int k0 = kk + 2 * half;
    // A fragment: 16x4, lane ln = row, holds K = k0, k0+1 (contiguous 8B load)
    v2f a;
    const float* ap = &x[(long long)(row0 + ln) * DIN + k0];
    a.x = ap[0];
    a.y = ap[1];
    #pragma unroll
    for (int c = 0; c < 8; ++c) {
      // B fragment: 4x16, lane ln = col, holds K = k0, k0+1
      v2f bf;
      bf.x = W[(long long)k0 * DOUT + c * 16 + ln];
      bf.y = W[(long long)(k0 + 1) * DOUT + c * 16 + ln];
      acc[c] = __builtin_amdgcn_wmma_f32_16x16x4_f32(
          /*neg_a=*/false, a, /*neg_b=*/false, bf,
          /*c_mod=*/(short)0, acc[c], /*reuse_a=*/false, /*reuse_b=*/false);
    }
  }

  #pragma unroll
  for (int c = 0; c < 8; ++c)
    #pragma unroll
    for (int i = 0; i < 8; ++i)
      h[(long long)(row0 + i + 8 * half) * DOUT + c * 16 + ln] = acc[c][i];
}

// ---------- per-row dot with a 128-vector: dst[row] = src[row,:] . vec ----------
__global__ void k_rowdot(const float* __restrict__ src, const float* __restrict__ vec,
                         float* __restrict__ dst, int rows) {
  const int wave = blockIdx.x * (blockDim.x >> 5) + (threadIdx.x >> 5);
  const int lane = threadIdx.x & 31;
  if (wave >= rows) return;
  const float* r = src + (long long)wave * DOUT;
  float s = r[lane]      * vec[lane]
          + r[lane + 32] * vec[lane + 32]
          + r[lane + 64] * vec[lane + 64]
          + r[lane + 96] * vec[lane + 96];
  #pragma unroll
  for (int off = 16; off > 0; off >>= 1) s += __shfl_xor(s, off, 32);
  if (lane == 0) dst[wave] = s;
}

// ---------- global max of logit e[m] = a_n[nidx[m]] (+ a_e[hidx[m]]) ----------
__global__ void k_logit_max(const float* __restrict__ a_n, const int* __restrict__ nidx,
                            const float* __restrict__ a_e, const int* __restrict__ hidx,
                            int M, unsigned* __restrict__ maxout) {
  __shared__ unsigned sm[256];
  long long tid = (long long)blockIdx.x * blockDim.x + threadIdx.x;
  long long stride = (long long)gridDim.x * blockDim.x;
  unsigned best = 0u;  // below ordf of every finite float
  for (long long m = tid; m < M; m += stride) {
    float e = a_n[nidx[m]];
    if (a_e) e += a_e[hidx[m]];
    unsigned o = ordf(e);
    best = (o > best) ? o : best;
  }
  sm[threadIdx.x] = best;
  __syncthreads();
  for (int s = 128; s > 0; s >>= 1) {
    if (threadIdx.x < (unsigned)s) {
      unsigned o = sm[threadIdx.x + s];
      if (o > sm[threadIdx.x]) sm[threadIdx.x] = o;
    }
    __syncthreads();
  }
  if (threadIdx.x == 0) atomicMax(maxout, sm[0]);
}

// ---------- global sum of exp(e[m] - gmax) ----------
__global__ void k_logit_sum(const float* __restrict__ a_n, const int* __restrict__ nidx,
                            const float* __restrict__ a_e, const int* __restrict__ hidx,
                            int M, const unsigned* __restrict__ maxin,
                            float* __restrict__ sumout) {
  __shared__ float sm[256];
  const float gmax = unordf(*maxin);
  long long tid = (long long)blockIdx.x * blockDim.x + threadIdx.x;
  long long stride = (long long)gridDim.x * blockDim.x;
  float acc = 0.0f;
  for (long long m = tid; m < M; m += stride) {
    float e = a_n[nidx[m]];
    if (a_e) e += a_e[hidx[m]];
    acc += __expf(e - gmax);
  }
  sm[threadIdx.x] = acc;
  __syncthreads();
  for (int s = 128; s > 0; s >>= 1) {
    if (threadIdx.x < (unsigned)s) sm[threadIdx.x] += sm[threadIdx.x + s];
    __syncthreads();
  }
  if (threadIdx.x == 0) atomicAdd(sumout, sm[0]);
}

// ---------- weighted scatter-add: out[dst[m],:] += feat[src[m],:] * softmax(e)[m] ----------
// 32 lanes per incidence, 4 floats per lane (float4 gather, 4 atomic f32 adds).
__global__ void k_scatter(const float* __restrict__ feat, const int* __restrict__ src_idx,
                          const int* __restrict__ dst_idx,
                          const float* __restrict__ a_n, const int* __restrict__ nidx,
                          const float* __restrict__ a_e, const int* __restrict__ hidx,
                          int M, const unsigned* __restrict__ maxin,
                          const float* __restrict__ sumin, float* __restrict__ out) {
  long long t = (long long)blockIdx.x * blockDim.x + threadIdx.x;
  const int m = (int)(t >> 5);
  if (m >= M) return;
  const int lane = (int)(t & 31);

  const float gmax = unordf(*maxin);
  const float gsum = *sumin;
  float e = a_n[nidx[m]];
  if (a_e) e += a_e[hidx[m]];
  const float w = __expf(e - gmax) / gsum;

  const int src = src_idx[m];
  const int dst = dst_idx[m];
  const float4 v = ((const float4*)(feat + (long long)src * DOUT))[lane];
  float* dp = out + (long long)dst * DOUT + lane * 4;
  atomicAdd(dp + 0, v.x * w);
  atomicAdd(dp + 1, v.y * w);
  atomicAdd(dp + 2, v.z * w);
  atomicAdd(dp + 3, v.w * w);
}

extern "C" void kernel_launch(void* const* d_in, const int* in_sizes, int n_in,
                              void* d_out, int out_size, void* d_ws, size_t ws_size,
                              hipStream_t stream) {
  const float* x         = (const float*)d_in[0];
  const int*   node_idx  = (const int*)d_in[1];
  const int*   he_idx    = (const int*)d_in[2];
  const float* W         = (const float*)d_in[3];
  const float* bias      = (const float*)d_in[4];
  const float* attn_node = (const float*)d_in[5];
  const float* attn_edge = (const float*)d_in[6];

  const int N = in_sizes[0] / DIN;
  const int M = in_sizes[1];
  const int H = HEDGES;
  float* out = (float*)d_out;

  // workspace layout (fp32): h[N*128] | a_n[N] | h_e[H*128] | a_e[H] | stats[4]
  float* ws  = (float*)d_ws;
  float* h   = ws;
  float* a_n = h + (size_t)N * DOUT;
  float* h_e = a_n + (((size_t)N + 3) & ~(size_t)3);
  float* a_e = h_e + (size_t)H * DOUT;
  unsigned* stats = (unsigned*)(a_e + (((size_t)H + 3) & ~(size_t)3));
  unsigned* max1 = stats + 0;  float* sum1 = (float*)(stats + 1);
  unsigned* max2 = stats + 2;  float* sum2 = (float*)(stats + 3);

  const int TB = 256;

  // 1) zero h_e and d_out, reset softmax stats
  k_init<<<1024, TB, 0, stream>>>(h_e, out, stats, (long long)H * DOUT, (long long)N * DOUT);

  // 2) h = x @ W + b  (WMMA fp32)
  {
    const int waves = N / 16;                 // N = 100000 -> 6250 tiles
    const int blocks = (waves + 7) / 8;       // 8 waves per 256-thread block
    k_gemm_wmma<<<blocks, TB, 0, stream>>>(x, W, bias, h, N);
  }

  // 3) a_n = h @ attn_node
  k_rowdot<<<(N + 7) / 8, TB, 0, stream>>>(h, attn_node, a_n, N);

  // 4-6) stage 1: global softmax over e1 = a_n[node_idx], scatter into h_e
  k_logit_max<<<1024, TB, 0, stream>>>(a_n, node_idx, nullptr, nullptr, M, max1);
  k_logit_sum<<<1024, TB, 0, stream>>>(a_n, node_idx, nullptr, nullptr, M, max1, sum1);
  {
    const long long threads = (long long)M * 32;
    k_scatter<<<(unsigned)((threads + TB - 1) / TB), TB, 0, stream>>>(
        h, node_idx, he_idx, a_n, node_idx, nullptr, nullptr, M, max1, sum1, h_e);
  }

  // 7) a_e = h_e @ attn_edge
  k_rowdot<<<(H + 7) / 8, TB, 0, stream>>>(h_e, attn_edge, a_e, H);

  // 8-10) stage 2: global softmax over e2 = a_e[he_idx] + a_n[node_idx], scatter into out
  k_logit_max<<<1024, TB, 0, stream>>>(a_n, node_idx, a_e, he_idx, M, max2);
  k_logit_sum<<<1024, TB, 0, stream>>>(a_n, node_idx, a_e, he_idx, M, max2, sum2);
  {
    const long long threads = (long long)M * 32;
    k_scatter<<<(unsigned)((threads + TB - 1) / TB), TB, 0, stream>>>(
        h_e, he_idx, node_idx, a_n, node_idx, a_e, he_idx, M, max2, sum2, out);
  }
}